// LocalPosUpdate_83786222011178
// MI455X (gfx1250) — compile-verified
//
#include <hip/hip_runtime.h>
#include <math.h>

// ---------------------------------------------------------------------------
// LocalPosUpdate for MI455X (gfx1250, wave32).
// Four fused 2-layer MLPs implemented with V_WMMA_F32_16X16X4_F32 tiles,
// hidden activations staged in LDS, gathers folded into A-fragment loads.
// ---------------------------------------------------------------------------

typedef __attribute__((ext_vector_type(2))) float v2f;
typedef __attribute__((ext_vector_type(8))) float v8f;

#define CUTOFF_F 3.5f
#define NG_ 16

// ---------------------------------------------------------------------------
__global__ void zero_kernel(float* __restrict__ p, int n) {
  int i = blockIdx.x * blockDim.x + threadIdx.x;
  if (i < n) p[i] = 0.0f;
}

// ---------------------------------------------------------------------------
// Edge geometry: unit vectors + Gaussian-smeared distances, and assemble the
// dist-MLP input X_dist[E,80] = [h_edge(64) | gauss(16)].
__global__ void edge_geom_kernel(const float* __restrict__ node_pos,
                                 const float* __restrict__ h_edge,
                                 const int* __restrict__ edge_index,
                                 float* __restrict__ unitv,   // [Epad,3]
                                 float* __restrict__ Xd,      // [Epad,80]
                                 int E) {
  int e = blockIdx.x * blockDim.x + threadIdx.x;
  if (e >= E) return;
  int s = edge_index[e];
  int d = edge_index[E + e];
  float vx = node_pos[3 * s + 0] - node_pos[3 * d + 0];
  float vy = node_pos[3 * s + 1] - node_pos[3 * d + 1];
  float vz = node_pos[3 * s + 2] - node_pos[3 * d + 2];
  float dist = sqrtf(vx * vx + vy * vy + vz * vz);
  float inv = 1.0f / (dist + 1e-8f);
  unitv[3 * e + 0] = vx * inv;
  unitv[3 * e + 1] = vy * inv;
  unitv[3 * e + 2] = vz * inv;

  float* row = Xd + (size_t)e * 80;
  const float4* he4 = (const float4*)(h_edge + (size_t)e * 64);
  float4* row4 = (float4*)row;
#pragma unroll
  for (int c = 0; c < 16; ++c) row4[c] = he4[c];

  const float dlt = CUTOFF_F / (float)(NG_ - 1);
  const float coeff = -0.5f / (dlt * dlt);
#pragma unroll
  for (int g = 0; g < NG_; ++g) {
    float u = dist - (float)g * dlt;
    row[64 + g] = __expf(coeff * u * u);
  }
}

// ---------------------------------------------------------------------------
// Fused 2-layer MLP: OUT = relu(X @ W1 + b1) @ W2 + b2, 16 rows per block.
// MODE 0: dense X [M,K1]
// MODE 1: X row t = concat(h_node[i0[3t+0..2]]) (128-wide segments, K1=384)
// MODE 2: X row t = concat(X[i0[t]], X[i1[t]], X[i2[t]]) (64-wide, K1=192)
// MODE 3: X row t = concat(X[t,:128], X2[t,:64]) (K1=192)
// NW waves compute the H-wide hidden (H = NW*16); NW2 waves compute layer 2.
template <int MODE, int K1, int H, int NOUT, int NW, int NW2>
__global__ __launch_bounds__(NW * 32) void mlp_fused_kernel(
    const float* __restrict__ X, const float* __restrict__ X2,
    const int* __restrict__ i0, const int* __restrict__ i1,
    const int* __restrict__ i2, const float* __restrict__ W1,
    const float* __restrict__ b1, const float* __restrict__ W2,
    const float* __restrict__ b2, float* __restrict__ OUT, int Mreal) {
  extern __shared__ float hid[];  // [16][H] hidden tile

  __builtin_prefetch(W1, 0, 0);   // global_prefetch_b8
  __builtin_prefetch(W2, 0, 0);

  const int lane = threadIdx.x & 31;
  const int wv = threadIdx.x >> 5;
  const int m0 = blockIdx.x * 16;
  const int mrow = lane & 15;                 // row (A) / col (B,C) index
  const int khalf = (lane >> 4) << 1;         // K sub-offset: 0 or 2
  const int r = m0 + mrow;                    // global row for A fragments
  const int tclamp = (r < Mreal) ? r : (Mreal - 1);
  const int ncol = wv * 16 + mrow;            // hidden column this lane owns

  // Pre-resolve gather bases for this lane's row.
  size_t base0 = 0, base1 = 0, base2 = 0;
  if constexpr (MODE == 1) {
    base0 = (size_t)i0[3 * tclamp + 0] * 128;
    base1 = (size_t)i0[3 * tclamp + 1] * 128;
    base2 = (size_t)i0[3 * tclamp + 2] * 128;
  } else if constexpr (MODE == 2) {
    base0 = (size_t)i0[tclamp] * 64;
    base1 = (size_t)i1[tclamp] * 64;
    base2 = (size_t)i2[tclamp] * 64;
  }

  // ---- Layer 1: hidden = relu(X @ W1 + b1) ----
  v8f acc = {};
  for (int kk = 0; kk < K1; kk += 4) {
    int kf = kk + khalf;
    v2f a;
    if constexpr (MODE == 0) {
      a = *(const v2f*)(X + (size_t)r * K1 + kf);
    } else if constexpr (MODE == 1) {
      int sg = kf >> 7;
      size_t b = (sg == 0) ? base0 : ((sg == 1) ? base1 : base2);
      a = *(const v2f*)(X + b + (kf & 127));
    } else if constexpr (MODE == 2) {
      int sg = kf >> 6;
      size_t b = (sg == 0) ? base0 : ((sg == 1) ? base1 : base2);
      a = *(const v2f*)(X + b + (kf & 63));
    } else {
      a = (kf < 128) ? *(const v2f*)(X + (size_t)r * 128 + kf)
                     : *(const v2f*)(X2 + (size_t)r * 64 + (kf - 128));
    }
    v2f bf;
    bf.x = W1[(size_t)kf * H + ncol];
    bf.y = W1[(size_t)(kf + 1) * H + ncol];
    acc = __builtin_amdgcn_wmma_f32_16x16x4_f32(false, a, false, bf, (short)0,
                                                acc, false, false);
  }

  // bias + relu -> LDS hidden tile (C layout: VGPR v -> M = v + 8*(lane>=16))
  {
    float bv = b1[ncol];
    int mb = (lane >> 4) * 8;
#pragma unroll
    for (int v = 0; v < 8; ++v) {
      float x = acc[v] + bv;
      hid[(size_t)(mb + v) * H + ncol] = (x > 0.0f) ? x : 0.0f;
    }
  }
  __syncthreads();

  // ---- Layer 2: OUT = hidden @ W2 + b2 ----
  if (wv < NW2) {
    const bool colok = (ncol < NOUT);
    v8f acc2 = {};
    for (int kk = 0; kk < H; kk += 4) {
      int kf = kk + khalf;
      v2f a = *(const v2f*)(hid + (size_t)mrow * H + kf);
      v2f bf;
      bf.x = colok ? W2[(size_t)kf * NOUT + ncol] : 0.0f;
      bf.y = colok ? W2[(size_t)(kf + 1) * NOUT + ncol] : 0.0f;
      acc2 = __builtin_amdgcn_wmma_f32_16x16x4_f32(false, a, false, bf,
                                                   (short)0, acc2, false,
                                                   false);
    }
    if (colok) {
      float bv = b2[ncol];
      int mb = (lane >> 4) * 8;
#pragma unroll
      for (int v = 0; v < 8; ++v) {
        int row = m0 + mb + v;
        if (row < Mreal) OUT[(size_t)row * NOUT + ncol] = acc2[v] + bv;
      }
    }
  }
}

// ---------------------------------------------------------------------------
// delta = w0*v1 + w1*v2 + w2*(v1 x v2); segment-sum into out[node_triples[:,0]]
__global__ void delta_scatter_kernel(const float* __restrict__ wbuf,   // [T,3]
                                     const float* __restrict__ unitv,  // [E,3]
                                     const int* __restrict__ trip,     // [T,3]
                                     const int* __restrict__ e1,
                                     const int* __restrict__ e2,
                                     float* __restrict__ out, int T) {
  int t = blockIdx.x * blockDim.x + threadIdx.x;
  if (t >= T) return;
  float w0 = wbuf[3 * t + 0], w1 = wbuf[3 * t + 1], w2 = wbuf[3 * t + 2];
  int a = e1[t], b = e2[t];
  float v1x = unitv[3 * a + 0], v1y = unitv[3 * a + 1], v1z = unitv[3 * a + 2];
  float v2x = unitv[3 * b + 0], v2y = unitv[3 * b + 1], v2z = unitv[3 * b + 2];
  float cx = v1y * v2z - v1z * v2y;
  float cy = v1z * v2x - v1x * v2z;
  float cz = v1x * v2y - v1y * v2x;
  float dx = w0 * v1x + w1 * v2x + w2 * cx;
  float dy = w0 * v1y + w1 * v2y + w2 * cy;
  float dz = w0 * v1z + w1 * v2z + w2 * cz;
  int i = trip[3 * t];
  unsafeAtomicAdd(&out[3 * i + 0], dx);
  unsafeAtomicAdd(&out[3 * i + 1], dy);
  unsafeAtomicAdd(&out[3 * i + 2], dz);
}

// ---------------------------------------------------------------------------
extern "C" void kernel_launch(void* const* d_in, const int* in_sizes, int n_in,
                              void* d_out, int out_size, void* d_ws,
                              size_t ws_size, hipStream_t stream) {
  const float* h_node = (const float*)d_in[0];
  const float* node_pos = (const float*)d_in[1];
  const float* h_edge = (const float*)d_in[2];
  const int* edge_index = (const int*)d_in[3];
  const int* node_triples = (const int*)d_in[4];
  const int* e1 = (const int*)d_in[5];
  const int* e2 = (const int*)d_in[6];
  const int* e3 = (const int*)d_in[7];
  const float* dW1 = (const float*)d_in[8];
  const float* db1 = (const float*)d_in[9];
  const float* dW2 = (const float*)d_in[10];
  const float* db2 = (const float*)d_in[11];
  const float* nW1 = (const float*)d_in[12];
  const float* nb1 = (const float*)d_in[13];
  const float* nW2 = (const float*)d_in[14];
  const float* nb2 = (const float*)d_in[15];
  const float* eW1 = (const float*)d_in[16];
  const float* eb1 = (const float*)d_in[17];
  const float* eW2 = (const float*)d_in[18];
  const float* eb2 = (const float*)d_in[19];
  const float* wW1 = (const float*)d_in[20];
  const float* wb1 = (const float*)d_in[21];
  const float* wW2 = (const float*)d_in[22];
  const float* wb2 = (const float*)d_in[23];

  const int E = in_sizes[2] / 64;   // h_edge is [E,64]
  const int T = in_sizes[5];        // e1 is [T]
  const int N = in_sizes[0] / 128;  // h_node is [N,128]
  const size_t Epad = (size_t)((E + 15) / 16) * 16;
  const size_t Tpad = (size_t)((T + 15) / 16) * 16;

  // Workspace layout (floats). X_dist overlays the h_nt region (dead after
  // the dist MLP finishes, before the node MLP writes h_nt).
  float* f_ws = (float*)d_ws;
  float* unitv = f_ws;                        // Epad*3
  float* h_e = unitv + Epad * 3;              // Epad*64
  float* h_nt = h_e + Epad * 64;              // Tpad*128
  float* Xd = h_nt;                           // Epad*80 (overlay)
  float* h_et = h_nt + Tpad * 128;            // Tpad*64
  float* wbuf = h_et + Tpad * 64;             // Tpad*3

  // 1) zero the output accumulator
  zero_kernel<<<(out_size + 255) / 256, 256, 0, stream>>>((float*)d_out,
                                                          out_size);
  // 2) edge geometry + dist-MLP input assembly
  edge_geom_kernel<<<(E + 255) / 256, 256, 0, stream>>>(node_pos, h_edge,
                                                        edge_index, unitv, Xd,
                                                        E);
  // 3) dist MLP: [E,80] -> 64 -> 64
  mlp_fused_kernel<0, 80, 64, 64, 4, 4>
      <<<(int)(Epad / 16), 128, 16 * 64 * sizeof(float), stream>>>(
          Xd, nullptr, nullptr, nullptr, nullptr, dW1, db1, dW2, db2, h_e, E);
  // 4) node MLP: gather triples of h_node, [T,384] -> 128 -> 128
  mlp_fused_kernel<1, 384, 128, 128, 8, 8>
      <<<(int)(Tpad / 16), 256, 16 * 128 * sizeof(float), stream>>>(
          h_node, nullptr, node_triples, nullptr, nullptr, nW1, nb1, nW2, nb2,
          h_nt, T);
  // 5) edge MLP: gather h_e[e1|e2|e3], [T,192] -> 64 -> 64
  mlp_fused_kernel<2, 192, 64, 64, 4, 4>
      <<<(int)(Tpad / 16), 128, 16 * 64 * sizeof(float), stream>>>(
          h_e, nullptr, e1, e2, e3, eW1, eb1, eW2, eb2, h_et, T);
  // 6) weight MLP: [h_nt|h_et] = [T,192] -> 128 -> 3
  mlp_fused_kernel<3, 192, 128, 3, 8, 1>
      <<<(int)(Tpad / 16), 256, 16 * 128 * sizeof(float), stream>>>(
          h_nt, h_et, nullptr, nullptr, nullptr, wW1, wb1, wW2, wb2, wbuf, T);
  // 7) geometric contraction + scatter-add
  delta_scatter_kernel<<<(T + 255) / 256, 256, 0, stream>>>(
      wbuf, unitv, node_triples, e1, e2, (float*)d_out, T);
}